// PagedAttention_31318901522439
// MI455X (gfx1250) — compile-verified
//
#include <hip/hip_runtime.h>

typedef __attribute__((ext_vector_type(2))) float v2f;
typedef __attribute__((ext_vector_type(8))) float v8f;

#define NUM_SEQS 32
#define NUM_HEADS 32
#define NUM_KV_HEADS 8
#define QPK 4            // GQA group size (queries per kv head)
#define HEAD_SIZE 128
#define BLOCK_SIZE 16
#define MAX_BLOCKS_PER_SEQ 256
#define NW 4             // waves per workgroup (wave32)
#define LDS_OSTR 17      // padded stride for merge scratch (bank-conflict dodge)
#define ATTN_SCALE 0.08838834764831845f  // 1/sqrt(128)

#define PAGE_FLOATS (HEAD_SIZE * BLOCK_SIZE)       // 2048 floats = 8KB (one K or V page)
#define PAGE_BYTES  (PAGE_FLOATS * 4)              // 8KB
#define PAIR_BYTES  (2 * PAGE_BYTES)               // 16KB (K page + V page)
#define LDS_BYTES   (NW * 2 * PAIR_BYTES)          // 128KB double-buffered pages

// Hard scheduling fence: nothing may be reordered across it by the machine scheduler.
#if __has_builtin(__builtin_amdgcn_sched_barrier)
#define SCHED_FENCE() __builtin_amdgcn_sched_barrier(0)
#else
#define SCHED_FENCE() asm volatile("" ::: "memory")
#endif

// Async-copy one 8KB page (K or V) into LDS: 16 x b128 per lane = 16 x 512B.
// lds/g already include lane*16. Offsets are immediates (apply to both addresses).
__device__ __forceinline__ void fill_page8k(uint32_t lds, const char* g) {
#define ASY(OFF) asm volatile("global_load_async_to_lds_b128 %0, %1, off offset:" #OFF \
                              :: "v"(lds), "v"(g) : "memory")
    ASY(0);    ASY(512);  ASY(1024); ASY(1536);
    ASY(2048); ASY(2560); ASY(3072); ASY(3584);
    ASY(4096); ASY(4608); ASY(5120); ASY(5632);
    ASY(6144); ASY(6656); ASY(7168); ASY(7680);
#undef ASY
}

__global__ __launch_bounds__(NW * 32, 1)
void paged_attn_f32_wmma_async(const float* __restrict__ qry,
                               const float* __restrict__ kcache,
                               const float* __restrict__ vcache,
                               const int*   __restrict__ btab,
                               const int*   __restrict__ ctx_lens,
                               float*       __restrict__ out)
{
    extern __shared__ char smem[];   // [NW][2][16KB] pages; reused as merge scratch

    const int seq  = blockIdx.x;       // 0..31
    const int kvh  = blockIdx.y;       // 0..7
    const int tid  = threadIdx.x;
    const int wave = tid >> 5;
    const int lane = tid & 31;
    const int lq   = lane & 15;        // col index (query / token row, depending on tile)
    const int lh   = lane >> 4;        // wave half

    const int ctx     = ctx_lens[seq];
    const int nblocks = (ctx + BLOCK_SIZE - 1) / BLOCK_SIZE;

    // LDS byte address of the dynamic shared base (generic ptr low 32 bits = LDS addr).
    const uint32_t lds0 = (uint32_t)(uintptr_t)smem;

    // ---- Preload Q^T as WMMA B-tiles (32 chunks of 4 head-dims), SCALE folded in.
    // B tile (4x16): lane holds column q = lq, rows K = 2*lh, 2*lh+1.
    v2f qb[32];
    {
        const float* qbase = qry + ((size_t)seq * NUM_HEADS + (size_t)kvh * QPK) * HEAD_SIZE;
        #pragma unroll
        for (int c = 0; c < 32; ++c) {
            const int d = 4 * c + 2 * lh;
            float x0 = 0.f, x1 = 0.f;
            if (lq < QPK) {
                x0 = qbase[lq * HEAD_SIZE + d]     * ATTN_SCALE;
                x1 = qbase[lq * HEAD_SIZE + d + 1] * ATTN_SCALE;
            }
            qb[c].x = x0; qb[c].y = x1;
        }
    }

    // ---- O^T accumulator: 8 tiles of 16x16 (M = head-dim chunk, N = query)
    v8f o[8];
    #pragma unroll
    for (int n = 0; n < 8; ++n) o[n] = (v8f){0.f,0.f,0.f,0.f,0.f,0.f,0.f,0.f};
    float m_run = -1e30f;
    float l_run = 0.f;

    const int* btrow = btab + (size_t)seq * MAX_BLOCKS_PER_SEQ;

    // Current page-pair buffer byte offset within smem (toggles by ^PAIR_BYTES).
    uint32_t boff = (uint32_t)(wave * 2) * PAIR_BYTES;

    // Prime the current buffer with this wave's first page pair.
    if (wave < nblocks) {
        const int phys = btrow[wave];
        const size_t pg = ((size_t)phys * NUM_KV_HEADS + kvh) * (size_t)PAGE_FLOATS;
        const uint32_t dst = lds0 + boff + (uint32_t)(lane * 16);
        fill_page8k(dst,              (const char*)(kcache + pg) + lane * 16);
        fill_page8k(dst + PAGE_BYTES, (const char*)(vcache + pg) + lane * 16);
    }

    for (int bi = wave; bi < nblocks; bi += NW) {
        // Kick off the next page pair into the other buffer, then wait for current.
        const int bn = bi + NW;
        if (bn < nblocks) {
            const int physn = btrow[bn];
            const size_t pgn = ((size_t)physn * NUM_KV_HEADS + kvh) * (size_t)PAGE_FLOATS;
            const uint32_t dst = lds0 + (boff ^ PAIR_BYTES) + (uint32_t)(lane * 16);
            fill_page8k(dst,              (const char*)(kcache + pgn) + lane * 16);
            fill_page8k(dst + PAGE_BYTES, (const char*)(vcache + pgn) + lane * 16);
            asm volatile("s_wait_asynccnt 0x20" ::: "memory"); // current 32 done, next 32 in flight
        } else {
            asm volatile("s_wait_asynccnt 0x0" ::: "memory");
        }

        // Derive LDS pointers directly from smem (+integer offset) so address-space
        // inference keeps them in AS3 -> ds_load (NOT flat_load).
        const float* kbuf = (const float*)(smem + boff);
        const float* vbuf = (const float*)(smem + boff + PAGE_BYTES);

        // ===== S^T = K_blk(16x128) x Q^T(128x16), 32 x wmma_f32_16x16x4_f32 =====
        // Depth-2 register pipeline, pinned with scheduling fences:
        // groups g+1,g+2 of DS loads are in flight behind group g's 8 WMMAs
        // (wait collapses to dscnt<=16 instead of 0).
        v8f s = (v8f){0.f,0.f,0.f,0.f,0.f,0.f,0.f,0.f};
        {
            v2f ka[32];
            #pragma unroll
            for (int j = 0; j < 16; ++j) {          // prologue: groups 0,1
                const int d0 = 4 * j + 2 * lh;
                ka[j] = *(const v2f*)(kbuf + ((((d0 >> 3) * BLOCK_SIZE) + lq) << 3) + (d0 & 7));
            }
            SCHED_FENCE();
            #pragma unroll
            for (int g = 0; g < 4; ++g) {
                if (g < 2) {                         // issue group g+2
                    #pragma unroll
                    for (int j = 0; j < 8; ++j) {
                        const int c  = 8 * (g + 2) + j;
                        const int d0 = 4 * c + 2 * lh;
                        ka[c] = *(const v2f*)(kbuf + ((((d0 >> 3) * BLOCK_SIZE) + lq) << 3) + (d0 & 7));
                    }
                    SCHED_FENCE();
                }
                #pragma unroll
                for (int j = 0; j < 8; ++j)
                    s = __builtin_amdgcn_wmma_f32_16x16x4_f32(false, ka[8 * g + j], false,
                                                              qb[8 * g + j], (short)0, s,
                                                              false, false);
                SCHED_FENCE();
            }
        }

        // ===== online softmax over this block (column q = lq) =====
        const int t0 = bi * BLOCK_SIZE + 8 * lh;   // row r holds token t0 + r
        float p[8];
        float bm = -1e30f;
        #pragma unroll
        for (int r = 0; r < 8; ++r) {
            const float sv = (t0 + r < ctx) ? s[r] : -1e30f;
            p[r] = sv;
            bm = fmaxf(bm, sv);
        }
        bm = fmaxf(bm, __shfl_xor(bm, 16, 32));
        const float m_new = fmaxf(m_run, bm);
        const float corr  = __expf(m_run - m_new);
        float bsum = 0.f;
        #pragma unroll
        for (int r = 0; r < 8; ++r) { p[r] = __expf(p[r] - m_new); bsum += p[r]; }
        bsum += __shfl_xor(bsum, 16, 32);
        l_run = l_run * corr + bsum;
        m_run = m_new;
        #pragma unroll
        for (int n = 0; n < 8; ++n) o[n] = o[n] * corr;

        // ===== O^T += V^T(128x16) x P^T(16x16) =====
        // Build P^T B-tiles from D-layout regs (half-wave shuffle swap).
        float sp[8];
        #pragma unroll
        for (int r = 0; r < 8; ++r) sp[r] = __shfl_xor(p[r], 16, 32);
        v2f pB[4];
        pB[0].x = lh ? sp[2] : p[0];  pB[0].y = lh ? sp[3] : p[1];  // token rows 0..3
        pB[1].x = lh ? sp[6] : p[4];  pB[1].y = lh ? sp[7] : p[5];  // rows 4..7
        pB[2].x = lh ? p[2]  : sp[0]; pB[2].y = lh ? p[3]  : sp[1]; // rows 8..11
        pB[3].x = lh ? p[6]  : sp[4]; pB[3].y = lh ? p[7]  : sp[5]; // rows 12..15

        // Depth-2 pipeline for V tiles as well (4 groups of 8).
        {
            v2f va[32];
            #pragma unroll
            for (int g = 0; g < 2; ++g)              // prologue: token-chunks 0,1
                #pragma unroll
                for (int j = 0; j < 8; ++j)
                    va[8 * g + j] = *(const v2f*)(vbuf + (16 * j + lq) * BLOCK_SIZE + 4 * g + 2 * lh);
            SCHED_FENCE();
            #pragma unroll
            for (int k = 0; k < 4; ++k) {
                if (k < 2) {                         // issue token-chunk k+2
                    #pragma unroll
                    for (int j = 0; j < 8; ++j)
                        va[8 * (k + 2) + j] =
                            *(const v2f*)(vbuf + (16 * j + lq) * BLOCK_SIZE + 4 * (k + 2) + 2 * lh);
                    SCHED_FENCE();
                }
                #pragma unroll
                for (int j = 0; j < 8; ++j)
                    o[j] = __builtin_amdgcn_wmma_f32_16x16x4_f32(false, va[8 * k + j], false, pB[k],
                                                                 (short)0, o[j], false, false);
                SCHED_FENCE();
            }
        }
        boff ^= PAIR_BYTES;
    }

    // All async traffic retired before the page buffers are reused as merge scratch.
    asm volatile("s_wait_asynccnt 0x0" ::: "memory");
    __syncthreads();

    float* lds_O = (float*)smem;                        // [NW][128][LDS_OSTR]
    float* lds_m = lds_O + NW * HEAD_SIZE * LDS_OSTR;   // [NW][16]
    float* lds_l = lds_m + NW * 16;                     // [NW][16]

    if (lane < 16) {
        lds_m[wave * 16 + lq] = m_run;
        lds_l[wave * 16 + lq] = l_run;
    }
    #pragma unroll
    for (int n = 0; n < 8; ++n) {
        #pragma unroll
        for (int r = 0; r < 8; ++r) {
            const int d = 16 * n + r + 8 * lh;
            lds_O[(wave * HEAD_SIZE + d) * LDS_OSTR + lq] = o[n][r];
        }
    }
    __syncthreads();

    // ---- cross-wave flash merge + store: 4 queries x 128 dims = 512 outputs
    for (int e = tid; e < QPK * HEAD_SIZE; e += NW * 32) {
        const int qi = e >> 7;       // 0..3
        const int d  = e & 127;
        float mg = -1e30f;
        #pragma unroll
        for (int w = 0; w < NW; ++w) mg = fmaxf(mg, lds_m[w * 16 + qi]);
        float lsum = 0.f, val = 0.f;
        #pragma unroll
        for (int w = 0; w < NW; ++w) {
            const float cw = __expf(lds_m[w * 16 + qi] - mg);
            lsum += cw * lds_l[w * 16 + qi];
            val  += cw * lds_O[(w * HEAD_SIZE + d) * LDS_OSTR + qi];
        }
        out[((size_t)seq * NUM_HEADS + (size_t)kvh * QPK + qi) * HEAD_SIZE + d] = val / lsum;
    }
}

extern "C" void kernel_launch(void* const* d_in, const int* in_sizes, int n_in,
                              void* d_out, int out_size, void* d_ws, size_t ws_size,
                              hipStream_t stream) {
    const float* qry    = (const float*)d_in[0];
    const float* kcache = (const float*)d_in[1];
    const float* vcache = (const float*)d_in[2];
    const int*   btab   = (const int*)d_in[3];
    const int*   ctxl   = (const int*)d_in[4];
    float*       out    = (float*)d_out;

    dim3 grid(NUM_SEQS, NUM_KV_HEADS);
    dim3 block(NW * 32);
    paged_attn_f32_wmma_async<<<grid, block, LDS_BYTES, stream>>>(
        qry, kcache, vcache, btab, ctxl, out);
}